// GAT_16037407883898
// MI455X (gfx1250) — compile-verified
//
#include <hip/hip_runtime.h>
#include <hip/hip_bf16.h>

// CDNA5 (gfx1250) GAT forward: WMMA fp32 GEMMs (register-pipelined, pair-packed B)
// + atomic segment-softmax. Wave32; V_WMMA_F32_16X16X4_F32 layouts per ISA §7.12.2.

typedef __attribute__((ext_vector_type(2))) float v2f;
typedef __attribute__((ext_vector_type(8))) float v8f;

#define NEG_SLOPE 0.2f

// ---------------------------------------------------------------------------
// Repack B[K x N] -> Bp[K/2 x N x 2]: Bp[kp][n] = (B[2kp][n], B[2kp+1][n]).
// Makes each WMMA B-fragment a single b64 load.
// ---------------------------------------------------------------------------
__global__ void gat_pack_pairs(const float* __restrict__ B,
                               float2* __restrict__ Bp, int K, int N) {
  const int tid = blockIdx.x * blockDim.x + threadIdx.x;
  const int total = (K >> 1) * N;
  if (tid >= total) return;
  const int kp = tid / N;
  const int n  = tid - kp * N;
  float2 v;
  v.x = B[(size_t)(2 * kp)     * N + n];
  v.y = B[(size_t)(2 * kp + 1) * N + n];
  Bp[tid] = v;
}

// ---------------------------------------------------------------------------
// GEMM: C[Mrows x N] = A[Mrows x K] @ B[K x N], fp32 WMMA 16x16x4.
// B is pair-packed (see gat_pack_pairs). One wave -> 16 (M) x 64 (N) panel:
// 4 accumulators, A-frag reused 4x per K-step, register double-buffered so
// next k-step's 5 b64 loads are in flight while current WMMAs issue.
// Requires N % 64 == 0, K % 8 == 0.
// ---------------------------------------------------------------------------
__global__ void gat_wmma_gemm_f32(const float* __restrict__ A,
                                  const float2* __restrict__ Bq,
                                  float* __restrict__ C,
                                  int Mrows, int N, int K) {
  const int lane   = threadIdx.x & 31;
  const int wave   = blockIdx.x * (blockDim.x >> 5) + (threadIdx.x >> 5);
  const int chunks = N >> 6;                       // 64-wide N chunks per tile-row
  const int Mtiles = (Mrows + 15) >> 4;
  const int tileM  = wave / chunks;
  const int chunk  = wave - tileM * chunks;
  if (tileM >= Mtiles) return;                     // wave-uniform: EXEC stays all-1s

  const int half = lane >> 4;                      // 0: lanes 0-15, 1: lanes 16-31
  const int l15  = lane & 15;
  int m = tileM * 16 + l15;
  if (m >= Mrows) m = Mrows - 1;                   // clamp (tail-safe; exact here)
  const int nl = chunk * 64 + l15;

  v8f acc[4];
#pragma unroll
  for (int t = 0; t < 4; ++t)
#pragma unroll
    for (int v = 0; v < 8; ++v) acc[t][v] = 0.0f;

  const float* __restrict__ Arow = A + (size_t)m * K;

  v2f a_cur, b_cur[4];
  {
    const int ka = half * 2;                       // k0 = 0
    a_cur.x = Arow[ka];
    a_cur.y = Arow[ka + 1];
    const float2* __restrict__ bb = Bq + (size_t)half * N + nl;
#pragma unroll
    for (int t = 0; t < 4; ++t) {
      const float2 w = bb[t * 16];
      b_cur[t].x = w.x; b_cur[t].y = w.y;
    }
  }

  for (int k0 = 0; k0 < K; k0 += 4) {
    // Prefetch next k-step (wrap to 0 on last iteration: dummy, discarded).
    const int k4 = (k0 + 4 < K) ? (k0 + 4) : 0;
    v2f a_nxt, b_nxt[4];
    {
      const int ka = k4 + half * 2;
      a_nxt.x = Arow[ka];
      a_nxt.y = Arow[ka + 1];
      const float2* __restrict__ bb = Bq + (size_t)((k4 >> 1) + half) * N + nl;
#pragma unroll
      for (int t = 0; t < 4; ++t) {
        const float2 w = bb[t * 16];
        b_nxt[t].x = w.x; b_nxt[t].y = w.y;
      }
    }
#pragma unroll
    for (int t = 0; t < 4; ++t)
      acc[t] = __builtin_amdgcn_wmma_f32_16x16x4_f32(
          false, a_cur, false, b_cur[t], (short)0, acc[t], false, false);
    a_cur = a_nxt;
#pragma unroll
    for (int t = 0; t < 4; ++t) b_cur[t] = b_nxt[t];
  }

#pragma unroll
  for (int t = 0; t < 4; ++t)
#pragma unroll
    for (int v = 0; v < 8; ++v) {
      const int row = tileM * 16 + half * 8 + v;
      if (row < Mrows) C[(size_t)row * N + nl + t * 16] = acc[t][v];
    }
}

// ---------------------------------------------------------------------------
// alpha_src[n,h] = dot(h[n,h,:], a_src[h,:]);  alpha_dst likewise. C % 4 == 0.
// ---------------------------------------------------------------------------
__global__ void gat_alpha(const float* __restrict__ h,
                          const float* __restrict__ a_src,
                          const float* __restrict__ a_dst,
                          float* __restrict__ as_out,
                          float* __restrict__ ad_out,
                          int Nn, int H, int C) {
  const int tid = blockIdx.x * blockDim.x + threadIdx.x;
  if (tid >= Nn * H) return;
  const int n = tid / H, hh = tid - n * H;
  const float4* __restrict__ hp = (const float4*)(h + ((size_t)n * H + hh) * C);
  const float4* __restrict__ sp = (const float4*)(a_src + (size_t)hh * C);
  const float4* __restrict__ dp = (const float4*)(a_dst + (size_t)hh * C);
  float accs = 0.f, accd = 0.f;
  for (int c = 0; c < (C >> 2); ++c) {
    const float4 v = hp[c], s = sp[c], d = dp[c];
    accs += v.x * s.x + v.y * s.y + v.z * s.z + v.w * s.w;
    accd += v.x * d.x + v.y * d.y + v.z * d.z + v.w * d.w;
  }
  as_out[tid] = accs;
  ad_out[tid] = accd;
}

// Order-preserving float<->uint for atomicMax on floats (incl. negatives).
__device__ __forceinline__ unsigned f2ord(float f) {
  unsigned u = __float_as_uint(f);
  return (u & 0x80000000u) ? ~u : (u | 0x80000000u);
}
__device__ __forceinline__ float ord2f(unsigned u) {
  return (u & 0x80000000u) ? __uint_as_float(u & 0x7FFFFFFFu)
                           : __uint_as_float(~u);
}

// Pass 1: leaky-relu edge score; running per-(dst,head) max via atomicMax.
__global__ void gat_edge_max(const float* __restrict__ as,
                             const float* __restrict__ ad,
                             const int* __restrict__ src,
                             const int* __restrict__ dst,
                             float* __restrict__ ebuf,
                             unsigned* __restrict__ msort,
                             int E, int H) {
  const int tid = blockIdx.x * blockDim.x + threadIdx.x;
  if (tid >= E * H) return;
  const int e = tid / H, hh = tid - e * H;
  const int s = src[e], d = dst[e];
  float v = as[(size_t)s * H + hh] + ad[(size_t)d * H + hh];
  v = (v > 0.f) ? v : NEG_SLOPE * v;               // leaky_relu
  ebuf[tid] = v;
  atomicMax(&msort[(size_t)d * H + hh], f2ord(v));
}

// Pass 2: ex = exp(e - max); per-(dst,head) denominator via atomicAdd.
__global__ void gat_edge_exp(const int* __restrict__ dst,
                             float* __restrict__ ebuf,
                             const unsigned* __restrict__ msort,
                             float* __restrict__ denom,
                             int E, int H) {
  const int tid = blockIdx.x * blockDim.x + threadIdx.x;
  if (tid >= E * H) return;
  const int e = tid / H, hh = tid - e * H;
  const int d = dst[e];
  const float m = ord2f(msort[(size_t)d * H + hh]);
  const float ex = expf(ebuf[tid] - m);
  ebuf[tid] = ex;
  atomicAdd(&denom[(size_t)d * H + hh], ex);
}

// Pass 3: out[dst] += alpha * h[src].  float4 gather, 4 non-returning atomics.
// (H*C) must be a multiple of 4; threads-per-edge = H*C/4.
__global__ void gat_edge_scatter(const float* __restrict__ h,
                                 const float* __restrict__ ebuf,
                                 const float* __restrict__ denom,
                                 const int* __restrict__ src,
                                 const int* __restrict__ dst,
                                 float* __restrict__ out,
                                 int E, int H, int C) {
  const int qpe = (H * C) >> 2;                    // float4 quads per edge
  const long long tid = (long long)blockIdx.x * blockDim.x + threadIdx.x;
  if (tid >= (long long)E * qpe) return;
  const int e = (int)(tid / qpe);
  const int q = (int)(tid - (long long)e * qpe);
  const int hh = (q * 4) / C;
  const int s = src[e], d = dst[e];
  const float alpha = ebuf[(size_t)e * H + hh] /
                      (denom[(size_t)d * H + hh] + 1e-16f);
  const float4 v = *(const float4*)(h + (size_t)s * H * C + q * 4);
  float* op = out + (size_t)d * H * C + (size_t)q * 4;
  atomicAdd(op + 0, v.x * alpha);
  atomicAdd(op + 1, v.y * alpha);
  atomicAdd(op + 2, v.z * alpha);
  atomicAdd(op + 3, v.w * alpha);
}

// out[i] = elu(in[i] + b[i % D]), float4-vectorized; total % 4 == 0, D % 4 == 0.
__global__ void gat_bias_elu(const float* __restrict__ in,
                             const float* __restrict__ b,
                             float* __restrict__ out,
                             long long total4, int D4) {
  const long long tid = (long long)blockIdx.x * blockDim.x + threadIdx.x;
  if (tid >= total4) return;
  const float4 vi = ((const float4*)in)[tid];
  const float4 vb = ((const float4*)b)[(int)(tid % D4)];
  float4 r;
  r.x = vi.x + vb.x; r.y = vi.y + vb.y; r.z = vi.z + vb.z; r.w = vi.w + vb.w;
  r.x = (r.x > 0.f) ? r.x : (expf(r.x) - 1.f);
  r.y = (r.y > 0.f) ? r.y : (expf(r.y) - 1.f);
  r.z = (r.z > 0.f) ? r.z : (expf(r.z) - 1.f);
  r.w = (r.w > 0.f) ? r.w : (expf(r.w) - 1.f);
  ((float4*)out)[tid] = r;
}

// Final: y[n,:2] = elu(out2[n,:] + b2) @ Wfc + bfc
__global__ void gat_fc(const float* __restrict__ out2,
                       const float* __restrict__ b2,
                       const float* __restrict__ Wfc,
                       const float* __restrict__ bfc,
                       float* __restrict__ y, int Nn) {
  const int n = blockIdx.x * blockDim.x + threadIdx.x;
  if (n >= Nn) return;
  float a0 = bfc[0], a1 = bfc[1];
  const float4* __restrict__ row = (const float4*)(out2 + (size_t)n * 64);
  const float4* __restrict__ bb  = (const float4*)b2;
  for (int c4 = 0; c4 < 16; ++c4) {
    const float4 v = row[c4], vb = bb[c4];
    float e0 = v.x + vb.x, e1 = v.y + vb.y, e2 = v.z + vb.z, e3 = v.w + vb.w;
    e0 = (e0 > 0.f) ? e0 : (expf(e0) - 1.f);
    e1 = (e1 > 0.f) ? e1 : (expf(e1) - 1.f);
    e2 = (e2 > 0.f) ? e2 : (expf(e2) - 1.f);
    e3 = (e3 > 0.f) ? e3 : (expf(e3) - 1.f);
    const int c = c4 * 4;
    a0 += e0 * Wfc[(c + 0) * 2] + e1 * Wfc[(c + 1) * 2] +
          e2 * Wfc[(c + 2) * 2] + e3 * Wfc[(c + 3) * 2];
    a1 += e0 * Wfc[(c + 0) * 2 + 1] + e1 * Wfc[(c + 1) * 2 + 1] +
          e2 * Wfc[(c + 2) * 2 + 1] + e3 * Wfc[(c + 3) * 2 + 1];
  }
  y[(size_t)n * 2 + 0] = a0;
  y[(size_t)n * 2 + 1] = a1;
}

static inline long long cdivll(long long a, long long b) { return (a + b - 1) / b; }

extern "C" void kernel_launch(void* const* d_in, const int* in_sizes, int n_in,
                              void* d_out, int out_size, void* d_ws, size_t ws_size,
                              hipStream_t stream) {
  if (n_in < 12) return;
  const float* x    = (const float*)d_in[0];
  const int*   ei   = (const int*)d_in[1];          // (2,E) int32 (JAX x64 off)
  const float* W1   = (const float*)d_in[2];        // 128 x 512
  const float* aS1  = (const float*)d_in[3];        // 8 x 64
  const float* aD1  = (const float*)d_in[4];
  const float* b1   = (const float*)d_in[5];        // 512
  const float* W2   = (const float*)d_in[6];        // 512 x 64
  const float* aS2  = (const float*)d_in[7];        // 1 x 64
  const float* aD2  = (const float*)d_in[8];
  const float* b2   = (const float*)d_in[9];        // 64
  const float* Wfc  = (const float*)d_in[10];       // 64 x 2
  const float* bfc  = (const float*)d_in[11];       // 2
  float* y = (float*)d_out;
  (void)out_size; (void)ws_size;

  const int Nn = in_sizes[0] / 128;                 // 50000
  const int E  = in_sizes[1] / 2;                   // 800000
  const int* src = ei;
  const int* dst = ei + E;

  // Workspace carve-out (~272 MB of floats). All offsets 16B-aligned.
  float* p = (float*)d_ws;
  auto take = [&p](size_t n) { float* r = p; p += n; return r; };
  float*    h1   = take((size_t)Nn * 512);          // x @ W1
  float*    out1 = take((size_t)Nn * 512);          // layer-1 aggregation
  float*    alS1 = take((size_t)Nn * 8);
  float*    alD1 = take((size_t)Nn * 8);
  unsigned* m1   = (unsigned*)take((size_t)Nn * 8);
  float*    den1 = take((size_t)Nn * 8);
  float*    e1   = take((size_t)E * 8);
  float*    h2   = take((size_t)Nn * 64);           // x2 @ W2
  float*    out2 = take((size_t)Nn * 64);
  float*    alS2 = take((size_t)Nn);
  float*    alD2 = take((size_t)Nn);
  unsigned* m2   = (unsigned*)take((size_t)Nn);
  float*    den2 = take((size_t)Nn);
  float*    e2   = take((size_t)E);
  float2*   W1p  = (float2*)take((size_t)128 * 512);  // pair-packed W1
  float2*   W2p  = (float2*)take((size_t)512 * 64);   // pair-packed W2

  // Zero accumulators / maxima / denominators (graph-capture safe).
  hipMemsetAsync(out1, 0, (size_t)Nn * 512 * 4, stream);
  hipMemsetAsync(m1,   0, (size_t)Nn * 8 * 4,  stream);
  hipMemsetAsync(den1, 0, (size_t)Nn * 8 * 4,  stream);
  hipMemsetAsync(out2, 0, (size_t)Nn * 64 * 4, stream);
  hipMemsetAsync(m2,   0, (size_t)Nn * 4,      stream);
  hipMemsetAsync(den2, 0, (size_t)Nn * 4,      stream);

  const int BLK = 256;                              // 8 wave32s = 1 full WGP pass

  // Pair-pack both weight matrices up front.
  gat_pack_pairs<<<(unsigned)cdivll(64LL * 512, BLK), BLK, 0, stream>>>(
      W1, W1p, 128, 512);
  gat_pack_pairs<<<(unsigned)cdivll(256LL * 64, BLK), BLK, 0, stream>>>(
      W2, W2p, 512, 64);

  // ---------------- Layer 1 (H=8, C=64) ----------------
  {
    const long long Mtiles = cdivll(Nn, 16);        // 3125
    const long long waves  = Mtiles * (512 / 64);   // 25000
    gat_wmma_gemm_f32<<<(unsigned)cdivll(waves, 8), BLK, 0, stream>>>(
        x, W1p, h1, Nn, 512, 128);
  }
  gat_alpha<<<(unsigned)cdivll((long long)Nn * 8, BLK), BLK, 0, stream>>>(
      h1, aS1, aD1, alS1, alD1, Nn, 8, 64);
  gat_edge_max<<<(unsigned)cdivll((long long)E * 8, BLK), BLK, 0, stream>>>(
      alS1, alD1, src, dst, e1, m1, E, 8);
  gat_edge_exp<<<(unsigned)cdivll((long long)E * 8, BLK), BLK, 0, stream>>>(
      dst, e1, m1, den1, E, 8);
  gat_edge_scatter<<<(unsigned)cdivll((long long)E * 128, BLK), BLK, 0, stream>>>(
      h1, e1, den1, src, dst, out1, E, 8, 64);
  // x2 = elu(out1 + b1), written over h1 (no longer needed).
  gat_bias_elu<<<(unsigned)cdivll((long long)Nn * 128, BLK), BLK, 0, stream>>>(
      out1, b1, h1, (long long)Nn * 128, 128);

  // ---------------- Layer 2 (H=1, C=64) ----------------
  {
    const long long waves = cdivll(Nn, 16);         // 3125 (N=64 -> 1 chunk)
    gat_wmma_gemm_f32<<<(unsigned)cdivll(waves, 8), BLK, 0, stream>>>(
        h1, W2p, h2, Nn, 64, 512);
  }
  gat_alpha<<<(unsigned)cdivll((long long)Nn, BLK), BLK, 0, stream>>>(
      h2, aS2, aD2, alS2, alD2, Nn, 1, 64);
  gat_edge_max<<<(unsigned)cdivll((long long)E, BLK), BLK, 0, stream>>>(
      alS2, alD2, src, dst, e2, m2, E, 1);
  gat_edge_exp<<<(unsigned)cdivll((long long)E, BLK), BLK, 0, stream>>>(
      dst, e2, m2, den2, E, 1);
  gat_edge_scatter<<<(unsigned)cdivll((long long)E * 16, BLK), BLK, 0, stream>>>(
      h2, e2, den2, src, dst, out2, E, 1, 64);

  // ---------------- FC head (bias+elu fused) ----------------
  gat_fc<<<(unsigned)cdivll((long long)Nn, BLK), BLK, 0, stream>>>(
      out2, b2, Wfc, bfc, y, Nn);
}